// QuantumEntanglement_90177133347657
// MI455X (gfx1250) — compile-verified
//
#include <hip/hip_runtime.h>
#include <hip/hip_bf16.h>
#include <math.h>

typedef __attribute__((ext_vector_type(2))) float v2f;
typedef __attribute__((ext_vector_type(4))) float v4f;
typedef __attribute__((ext_vector_type(8))) float v8f;

#define NUM_EXPERTS 8
#define D_MODEL 512
#define BATCH 16384

__device__ __forceinline__ float mix_w(int i, int j, const float* __restrict__ C) {
    // W[i,j] = 1.0 on the diagonal, sigmoid(C[i,j]) off-diagonal
    float c = C[i * NUM_EXPERTS + j];
    float s = 1.0f / (1.0f + __expf(-c));
    return (i == j) ? 1.0f : s;
}

// Each wave handles one batch row b; loops over 32 tiles of 16 d-columns.
// Per tile: out^T(16d x 8e) = X^T(16d x 8j) * W^T(8j x 8e) via two
// v_wmma_f32_16x16x4_f32 accumulating K=4 each.
__global__ void __launch_bounds__(256)
entangle_wmma_kernel(const float* __restrict__ x,   // [B, E, D]
                     const float* __restrict__ C,   // [E, E]
                     float* __restrict__ out) {     // [B, E, D]
    const int lane  = threadIdx.x & 31;
    const int wave  = threadIdx.x >> 5;
    const int b     = blockIdx.x * (blockDim.x >> 5) + wave;   // batch row
    const int half  = lane >> 4;       // 0: lanes 0-15, 1: lanes 16-31
    const int lm    = lane & 15;       // M (d within tile) for A; N (expert) for B/D

    // ---- B operand: W^T, built once per wave (pure function of lane) ----
    // B[k, n] = W[n, jb + k]; lane holds n = lm (only n<8 meaningful),
    // VGPR0 -> k = 2*half, VGPR1 -> k = 2*half+1.
    const int n  = lm & 7;             // clamp so lanes with lm>=8 stay in-bounds
    const int jA = 2 * half;
    v2f bop0, bop1;
    bop0.x = mix_w(n, jA + 0, C);
    bop0.y = mix_w(n, jA + 1, C);
    bop1.x = mix_w(n, jA + 4, C);
    bop1.y = mix_w(n, jA + 5, C);

    const float* xb = x + (size_t)b * (NUM_EXPERTS * D_MODEL);

    #pragma unroll 4
    for (int d0 = 0; d0 < D_MODEL; d0 += 16) {
        // ---- A operand: X^T tile. A[m,k]: lane m = lm, VGPR0 -> j = jA,
        // VGPR1 -> j = jA+1 (step 0); step 1 adds 4 to j. ----
        const float* p0 = xb + jA * D_MODEL + d0 + lm;      // j = 2*half
        v2f a0, a1;
        a0.x = __builtin_nontemporal_load(p0);
        a0.y = __builtin_nontemporal_load(p0 + 1 * D_MODEL);
        a1.x = __builtin_nontemporal_load(p0 + 4 * D_MODEL);
        a1.y = __builtin_nontemporal_load(p0 + 5 * D_MODEL);

        v8f c = {};
        // D = A*B + C  (8 args: neg_a, A, neg_b, B, c_mod, C, reuse_a, reuse_b)
        c = __builtin_amdgcn_wmma_f32_16x16x4_f32(false, a0, false, bop0,
                                                  (short)0, c, false, false);
        c = __builtin_amdgcn_wmma_f32_16x16x4_f32(false, a1, false, bop1,
                                                  (short)0, c, false, false);

        // ---- Store: lane L (lm<8) holds out[b, e=lm, d0 + half*8 + r] in c[r]
        // -> two contiguous 16B stores per active lane (32B/lane, streaming). ----
        if (lm < 8) {
            float* po = out + ((size_t)b * NUM_EXPERTS + lm) * D_MODEL
                            + d0 + half * 8;
            v4f lo = {c[0], c[1], c[2], c[3]};
            v4f hi = {c[4], c[5], c[6], c[7]};
            __builtin_nontemporal_store(lo, (v4f*)po);
            __builtin_nontemporal_store(hi, (v4f*)(po + 4));
        }
    }
}

// Independent tiny kernel: entanglement measure from batch 0 only
// (reads 16 KB of input; one workgroup; deterministic reduction order).
__global__ void __launch_bounds__(256)
measure_kernel(const float* __restrict__ x,   // [B, E, D], uses batch 0
               const float* __restrict__ C,
               float* __restrict__ out_meas) {
    __shared__ float W[NUM_EXPERTS][NUM_EXPERTS];
    __shared__ float partial[256];
    const int t = threadIdx.x;

    if (t < NUM_EXPERTS * NUM_EXPERTS) {
        int i = t >> 3, j = t & 7;
        W[i][j] = mix_w(i, j, C);
    }
    __syncthreads();

    float acc = 0.0f;
    for (int idx = t; idx < NUM_EXPERTS * D_MODEL; idx += 256) {
        int i = idx >> 9;        // expert
        int d = idx & (D_MODEL - 1);
        float s = 0.0f;
        #pragma unroll
        for (int j = 0; j < NUM_EXPERTS; ++j)
            s += W[i][j] * x[j * D_MODEL + d];
        acc += s * s;
    }
    partial[t] = acc;
    __syncthreads();

    for (int stride = 128; stride > 0; stride >>= 1) {
        if (t < stride) partial[t] += partial[t + stride];
        __syncthreads();
    }
    if (t == 0)
        out_meas[0] = fabsf(partial[0]) /
                      ((float)(NUM_EXPERTS * D_MODEL) + 1e-8f);
}

extern "C" void kernel_launch(void* const* d_in, const int* in_sizes, int n_in,
                              void* d_out, int out_size, void* d_ws, size_t ws_size,
                              hipStream_t stream) {
    const float* x = (const float*)d_in[0];   // expert_states [16384, 8, 512] f32
    const float* C = (const float*)d_in[1];   // nonlocal_correlation [8, 8] f32
    float* out = (float*)d_out;               // [16384*8*512] entangled + [1] measure

    // 16384 waves (one per batch row), 8 waves per 256-thread block
    entangle_wmma_kernel<<<BATCH / 8, 256, 0, stream>>>(x, C, out);
    measure_kernel<<<1, 256, 0, stream>>>(x, C,
                                          out + (size_t)BATCH * NUM_EXPERTS * D_MODEL);
}